// GCN_88553635709104
// MI455X (gfx1250) — compile-verified
//
#include <hip/hip_runtime.h>

#define N_NODES 50000
#define N_EDGES 800000
#define D 128

typedef __attribute__((ext_vector_type(2))) float v2f;
typedef __attribute__((ext_vector_type(8))) float v8f;

// ---------------- utility: zero a float buffer (grid-stride) ----------------
__global__ void zero_kernel(float* __restrict__ p, int n) {
    int i = blockIdx.x * blockDim.x + threadIdx.x;
    int stride = gridDim.x * blockDim.x;
    for (; i < n; i += stride) p[i] = 0.0f;
}

// ---------------- out-degree count via f32 atomics ----------------
__global__ void deg_count_kernel(const int* __restrict__ src, float* __restrict__ deg) {
    int e = blockIdx.x * blockDim.x + threadIdx.x;
    if (e < N_EDGES) atomicAdd(&deg[src[e]], 1.0f);
}

// deg -> 1/max(deg,1) in place
__global__ void deg_inv_kernel(float* __restrict__ deg) {
    int i = blockIdx.x * blockDim.x + threadIdx.x;
    if (i < N_NODES) deg[i] = 1.0f / fmaxf(deg[i], 1.0f);
}

// ---------------- SpMM: agg[dst] += h[src] * deg_inv[src] ----------------
// One wave (32 lanes) per edge; each lane handles 4 consecutive floats.
// All of h (25.6MB) sits in L2, so gathers/atomics are L2-resident.
__global__ void spmm_kernel(const float* __restrict__ h,
                            const int* __restrict__ src,
                            const int* __restrict__ dst,
                            const float* __restrict__ deg_inv,
                            float* __restrict__ agg) {
    int idx  = blockIdx.x * blockDim.x + threadIdx.x;  // N_EDGES*32 threads exactly
    int e    = idx >> 5;
    int lane = idx & 31;
    int s = src[e];
    int d = dst[e];
    float sc = deg_inv[s];
    const float4 v = *(const float4*)(h + (long)s * D + lane * 4);
    float* o = agg + (long)d * D + lane * 4;
    atomicAdd(o + 0, v.x * sc);
    atomicAdd(o + 1, v.y * sc);
    atomicAdd(o + 2, v.z * sc);
    atomicAdd(o + 3, v.w * sc);
}

// ---------------- GEMM: out = act(A @ W + b) via V_WMMA_F32_16X16X4_F32 ----
// Block = 256 threads = 8 waves. blockIdx.x selects a 16-row strip of A;
// wave w owns output columns [16w, 16w+16). K=128 -> 32 unrolled WMMAs.
// A-frag (16x4 f32): lane m=lane&15 is row M; K-half = lane>>4 (K pair 0/1 vs 2/3).
// B-frag (4x16 f32): lane n=lane&15 is col N; same K-half split.
// C/D (16x16 f32, 8 VGPRs): VGPR r holds M = r + 8*(lane>>4), N = lane&15.
__global__ void gemm_wmma_kernel(const float* __restrict__ A,   // [N_NODES][128]
                                 const float* __restrict__ W,   // [128][128]
                                 const float* __restrict__ bias,// [128]
                                 float* __restrict__ out,       // [N_NODES][128]
                                 int relu) {
    const int row0  = blockIdx.x * 16;
    const int wave  = threadIdx.x >> 5;   // 0..7 -> column tile
    const int lane  = threadIdx.x & 31;
    const int m     = lane & 15;
    const int khalf = lane >> 4;          // 0 or 1
    const int col   = wave * 16 + m;

    const float* arow = A + (long)(row0 + m) * D;

    v8f c;
    const float bv = bias[col];           // bias depends only on N (col)
#pragma unroll
    for (int r = 0; r < 8; ++r) c[r] = bv;

#pragma unroll
    for (int ks = 0; ks < 32; ++ks) {
        const int k0 = ks * 4 + khalf * 2;          // even -> 8B-aligned
        v2f a = *(const v2f*)(arow + k0);           // A[M][k0], A[M][k0+1]
        v2f b;
        b.x = W[(long)k0 * D + col];                // W[k0][N]
        b.y = W[(long)(k0 + 1) * D + col];          // W[k0+1][N]
        c = __builtin_amdgcn_wmma_f32_16x16x4_f32(
                /*neg_a=*/false, a, /*neg_b=*/false, b,
                /*c_mod=*/(short)0, c, /*reuse_a=*/false, /*reuse_b=*/false);
    }

#pragma unroll
    for (int r = 0; r < 8; ++r) {
        float v = c[r];
        if (relu) v = fmaxf(v, 0.0f);
        out[(long)(row0 + khalf * 8 + r) * D + col] = v;
    }
}

// ---------------- column sums for mean readout ----------------
// 128 threads per block; thread t owns column t; block handles a row chunk.
__global__ void colsum_kernel(const float* __restrict__ h, float* __restrict__ colsum) {
    const int col = threadIdx.x;
    const int r0  = blockIdx.x * 256;
    const int r1  = min(r0 + 256, N_NODES);
    float s = 0.0f;
    for (int r = r0; r < r1; ++r) s += h[(long)r * D + col];
    atomicAdd(&colsum[col], s);
}

// ---------------- PI = h @ policy_w + policy_b (one wave per node) ---------
__global__ void pi_kernel(const float* __restrict__ h,
                          const float* __restrict__ pw,
                          const float* __restrict__ pb,
                          float* __restrict__ out) {
    int idx  = blockIdx.x * blockDim.x + threadIdx.x;  // N_NODES*32 exactly
    int node = idx >> 5;
    int lane = idx & 31;
    const float4 hv = *(const float4*)(h  + (long)node * D + lane * 4);
    const float4 wv = *(const float4*)(pw + lane * 4);
    float s = hv.x * wv.x + hv.y * wv.y + hv.z * wv.z + hv.w * wv.w;
#pragma unroll
    for (int off = 16; off; off >>= 1) s += __shfl_xor(s, off, 32);
    if (lane == 0) out[node] = s + pb[0];
}

// ---------------- V = mean(h) @ value_w + value_b (single wave) ------------
__global__ void v_kernel(const float* __restrict__ colsum,
                         const float* __restrict__ vw,
                         const float* __restrict__ vb,
                         float* __restrict__ out) {
    int lane = threadIdx.x;  // 32 threads
    const float4 cv = *(const float4*)(colsum + lane * 4);
    const float4 wv = *(const float4*)(vw + lane * 4);
    const float inv = 1.0f / (float)N_NODES;
    float s = (cv.x * wv.x + cv.y * wv.y + cv.z * wv.z + cv.w * wv.w) * inv;
#pragma unroll
    for (int off = 16; off; off >>= 1) s += __shfl_xor(s, off, 32);
    if (lane == 0) out[0] = s + vb[0];
}

extern "C" void kernel_launch(void* const* d_in, const int* in_sizes, int n_in,
                              void* d_out, int out_size, void* d_ws, size_t ws_size,
                              hipStream_t stream) {
    const float* features = (const float*)d_in[0];
    const int*   src      = (const int*)  d_in[1];
    const int*   dst      = (const int*)  d_in[2];
    const float* W[3]     = {(const float*)d_in[3], (const float*)d_in[5], (const float*)d_in[7]};
    const float* b[3]     = {(const float*)d_in[4], (const float*)d_in[6], (const float*)d_in[8]};
    const float* pw       = (const float*)d_in[9];
    const float* pb       = (const float*)d_in[10];
    const float* vw       = (const float*)d_in[11];
    const float* vb       = (const float*)d_in[12];
    float* out = (float*)d_out;

    // Workspace layout (floats): [deg/deg_inv 50176][colsum 128][agg 6.4M][h 6.4M]
    float* deg    = (float*)d_ws;
    float* colsum = deg + 50176;
    float* agg    = colsum + 128;
    float* h      = agg + (long)N_NODES * D;

    // Zero deg + colsum (contiguous 50304 floats); fresh every call -> deterministic.
    zero_kernel<<<256, 256, 0, stream>>>(deg, 50176 + 128);

    // Out-degree, then invert in place.
    deg_count_kernel<<<N_EDGES / 256, 256, 0, stream>>>(src, deg);
    deg_inv_kernel<<<(N_NODES + 255) / 256, 256, 0, stream>>>(deg);

    const float* hin = features;
    for (int l = 0; l < 3; ++l) {
        zero_kernel<<<2048, 256, 0, stream>>>(agg, N_NODES * D);
        spmm_kernel<<<(N_EDGES * 32) / 256, 256, 0, stream>>>(hin, src, dst, deg, agg);
        gemm_wmma_kernel<<<N_NODES / 16, 256, 0, stream>>>(agg, W[l], b[l], h, (l < 2) ? 1 : 0);
        hin = h;
    }

    colsum_kernel<<<(N_NODES + 255) / 256, 128, 0, stream>>>(h, colsum);
    pi_kernel<<<(N_NODES * 32) / 256, 256, 0, stream>>>(h, pw, pb, out);
    v_kernel<<<1, 32, 0, stream>>>(colsum, vw, vb, out + N_NODES);
}